// PointConvNN_49701361550139
// MI455X (gfx1250) — compile-verified
//
#include <hip/hip_runtime.h>
#include <stdint.h>

#define BB 4
#define NN 8192

typedef float v2f __attribute__((ext_vector_type(2)));
typedef float v8f __attribute__((ext_vector_type(8)));

// ---------------- split x(B,6,N) -> p(B,N,3), f(B,N,3) ----------------
__global__ void k_split(const float* __restrict__ x, float* __restrict__ p, float* __restrict__ f) {
  int i = blockIdx.x * blockDim.x + threadIdx.x;
  if (i >= BB * NN) return;
  int b = i / NN, n = i % NN;
  const float* xb = x + (size_t)b * 6 * NN;
#pragma unroll
  for (int c = 0; c < 3; c++) {
    p[(size_t)i * 3 + c] = xb[(size_t)c * NN + n];
    f[(size_t)i * 3 + c] = xb[(size_t)(3 + c) * NN + n];
  }
}

// ---------------- brute-force kNN (stable ties, ascending d2) ----------------
template <int KK>
__global__ void k_knn(const float* __restrict__ p, int qAbs, int Mq, int pAbs, int Np,
                      float radius2, int* __restrict__ idx, float* __restrict__ d2out) {
  int t = blockIdx.x * blockDim.x + threadIdx.x;
  if (t >= BB * Mq) return;
  int b = t / Mq, m = t % Mq;
  const float* pb = p + (size_t)b * NN * 3;
  float qx = pb[(size_t)m * qAbs * 3 + 0];
  float qy = pb[(size_t)m * qAbs * 3 + 1];
  float qz = pb[(size_t)m * qAbs * 3 + 2];
  float bd[KK]; int bi[KK];
#pragma unroll
  for (int k = 0; k < KK; k++) { bd[k] = 3.4e38f; bi[k] = 0; }
  for (int j = 0; j < Np; j++) {
    float dx = pb[(size_t)j * pAbs * 3 + 0] - qx;
    float dy = pb[(size_t)j * pAbs * 3 + 1] - qy;
    float dz = pb[(size_t)j * pAbs * 3 + 2] - qz;
    float d2 = dx * dx + dy * dy + dz * dz;
    if (d2 < bd[KK - 1]) {
      int pos = KK - 1;
      while (pos > 0 && bd[pos - 1] > d2) { bd[pos] = bd[pos - 1]; bi[pos] = bi[pos - 1]; pos--; }
      bd[pos] = d2; bi[pos] = j;
    }
  }
  int i0 = bi[0];
  for (int k = 0; k < KK; k++) {
    int v = bi[k];
    if (radius2 >= 0.f && bd[k] > radius2) v = i0;
    idx[(size_t)t * KK + k] = v;
    if (d2out) d2out[(size_t)t * KK + k] = bd[k];
  }
}

// ---------------- conv H = [rel(3) | feat[idx]] padded ----------------
__global__ void k_build_conv_h(const float* __restrict__ p, const float* __restrict__ feat,
                               int featC, int Np, int qAbs, int Mq, int pAbs,
                               const int* __restrict__ idx, int K,
                               float* __restrict__ H, int cinPad) {
  size_t i = (size_t)blockIdx.x * blockDim.x + threadIdx.x;
  size_t rows = (size_t)BB * Mq * K;
  if (i >= rows * (size_t)cinPad) return;
  size_t r = i / cinPad; int c = (int)(i % cinPad);
  size_t bm = r / K;
  int m = (int)(bm % Mq);
  int b = (int)(bm / Mq);
  int j = idx[r];
  float v = 0.f;
  if (c < 3) {
    const float* pb = p + (size_t)b * NN * 3;
    v = pb[(size_t)j * pAbs * 3 + c] - pb[(size_t)m * qAbs * 3 + c];
  } else if (c < 3 + featC) {
    v = feat[((size_t)b * Np + j) * featC + (c - 3)];
  }
  H[i] = v;
}

// ---------------- deconv H = [interp(fc) | fd] padded ----------------
__global__ void k_build_deconv_h(const float* __restrict__ fc, int Cc, int Mc,
                                 const float* __restrict__ fd, int Cd, int Md,
                                 const int* __restrict__ idx, const float* __restrict__ d2,
                                 float* __restrict__ H, int cinPad) {
  const int K = 16;
  size_t i = (size_t)blockIdx.x * blockDim.x + threadIdx.x;
  size_t rows = (size_t)BB * Md;
  if (i >= rows * (size_t)cinPad) return;
  size_t r = i / cinPad; int c = (int)(i % cinPad);
  int b = (int)(r / Md);
  float v = 0.f;
  if (c < Cc) {
    const float* dr = d2 + r * K;
    const int* ir = idx + r * K;
    float w[K]; float wsum = 0.f;
#pragma unroll
    for (int k = 0; k < K; k++) {
      float dd = dr[k]; dd = dd > 0.f ? dd : 0.f;
      w[k] = 1.f / (dd + 1e-8f); wsum += w[k];
    }
    float acc = 0.f;
#pragma unroll
    for (int k = 0; k < K; k++)
      acc += w[k] * fc[((size_t)b * Mc + ir[k]) * Cc + c];
    v = acc / wsum;
  } else if (c < Cc + Cd) {
    v = fd[r * Cd + (c - Cc)];
  }
  H[i] = v;
}

// ---------------- pack W (cout,cin) -> zero-padded (coutPad,cinPad) ----------------
__global__ void k_pack_w(const float* __restrict__ W, float* __restrict__ Wp,
                         int cout, int cin, int coutPad, int cinPad) {
  size_t i = (size_t)blockIdx.x * blockDim.x + threadIdx.x;
  if (i >= (size_t)coutPad * cinPad) return;
  int o = (int)(i / cinPad), c = (int)(i % cinPad);
  Wp[i] = (o < cout && c < cin) ? W[(size_t)o * cin + c] : 0.f;
}

// ---------------- fp32 WMMA GEMM: Y[rows,coutPad] = A[rows,cinPad] * Wp^T ----------------
// Each wave computes a 16x32 output strip (two N-tiles) so one A fragment feeds
// two v_wmma_f32_16x16x4_f32 per k-step (3 loads -> 2 wmma).
__global__ void k_gemm_wmma(const float* __restrict__ A, const float* __restrict__ Wp,
                            float* __restrict__ Y, int rows, int cinPad, int coutPad) {
  int tile = blockIdx.x * blockDim.y + threadIdx.y;
  int tn = coutPad >> 5;                 // N strips of 32
  int ntiles = (rows >> 4) * tn;
  if (tile >= ntiles) return;
  int tm = tile / tn, tc = tile % tn;
  int lane = threadIdx.x;
  int half = lane >> 4, lid = lane & 15;
  int m0 = tm * 16, n0 = tc * 32;
  v8f acc0 = {0.f, 0.f, 0.f, 0.f, 0.f, 0.f, 0.f, 0.f};
  v8f acc1 = {0.f, 0.f, 0.f, 0.f, 0.f, 0.f, 0.f, 0.f};
  // A 16x4 f32: lanes0-15 M=lid K={k0,k0+1}; lanes16-31 M=lid K={k0+2,k0+3}
  const float* arow = A + (size_t)(m0 + lid) * cinPad + half * 2;
  // B 4x16 f32 = Wp^T tile: lanes0-15 N=lid K={k0,k0+1}; lanes16-31 N=lid K={k0+2,k0+3}
  const float* brow0 = Wp + (size_t)(n0 + lid) * cinPad + half * 2;
  const float* brow1 = Wp + (size_t)(n0 + 16 + lid) * cinPad + half * 2;
  for (int k0 = 0; k0 < cinPad; k0 += 4) {
    v2f a, b0, b1;
    a.x = arow[k0];  a.y = arow[k0 + 1];
    b0.x = brow0[k0]; b0.y = brow0[k0 + 1];
    b1.x = brow1[k0]; b1.y = brow1[k0 + 1];
    acc0 = __builtin_amdgcn_wmma_f32_16x16x4_f32(false, a, false, b0, (short)0, acc0, false, false);
    acc1 = __builtin_amdgcn_wmma_f32_16x16x4_f32(false, a, false, b1, (short)0, acc1, false, false);
  }
  // D 16x16 f32: VGPR r -> M=r (lanes0-15) / M=8+r (lanes16-31), N=lid
#pragma unroll
  for (int r = 0; r < 8; r++) {
    int m = m0 + half * 8 + r;
    float* yrow = Y + (size_t)m * coutPad + n0 + lid;
    yrow[0]  = acc0[r];
    yrow[16] = acc1[r];
  }
}

// ---------------- BN stats: per-channel sum & sumsq (LDS partials + global atomics) ----------------
__global__ void k_zero(float* __restrict__ p, int n) {
  int i = blockIdx.x * blockDim.x + threadIdx.x;
  if (i < n) p[i] = 0.f;
}

__global__ void k_reduce_stats(const float* __restrict__ Y, int rows, int cout,
                               float* __restrict__ stats) {
  __shared__ float sh[1024];
  for (int c = threadIdx.x; c < 2 * cout; c += blockDim.x) sh[c] = 0.f;
  __syncthreads();
  size_t total = (size_t)rows * cout;
  for (size_t i = (size_t)blockIdx.x * blockDim.x + threadIdx.x; i < total;
       i += (size_t)gridDim.x * blockDim.x) {
    float v = Y[i]; int c = (int)(i % cout);
    atomicAdd(&sh[c], v);
    atomicAdd(&sh[cout + c], v * v);
  }
  __syncthreads();
  for (int c = threadIdx.x; c < 2 * cout; c += blockDim.x)
    atomicAdd(&stats[c], sh[c]);
}

__global__ void k_bn_relu(float* __restrict__ Y, const float* __restrict__ stats,
                          const float* __restrict__ g, const float* __restrict__ b,
                          int rows, int cout) {
  size_t i = (size_t)blockIdx.x * blockDim.x + threadIdx.x;
  if (i >= (size_t)rows * cout) return;
  int c = (int)(i % cout);
  float inv = 1.f / (float)rows;
  float mean = stats[c] * inv;
  float var = stats[cout + c] * inv - mean * mean;
  float y = g[c] * (Y[i] - mean) * rsqrtf(var + 1e-5f) + b[c];
  Y[i] = y > 0.f ? y : 0.f;
}

// ---------------- max over K neighbors ----------------
__global__ void k_maxpool(const float* __restrict__ Y, float* __restrict__ F,
                          int BM, int K, int C) {
  size_t i = (size_t)blockIdx.x * blockDim.x + threadIdx.x;
  if (i >= (size_t)BM * C) return;
  size_t g = i / C; int c = (int)(i % C);
  float m = -3.4e38f;
  for (int k = 0; k < K; k++) {
    float v = Y[((size_t)g * K + k) * C + c];
    m = v > m ? v : m;
  }
  F[i] = m;
}

// ---------------- final fc output: out[b,c,n] = Y[b,n,c] + bias[c] ----------------
__global__ void k_fc_out(const float* __restrict__ Y, const float* __restrict__ bias,
                         float* __restrict__ out) {
  size_t i = (size_t)blockIdx.x * blockDim.x + threadIdx.x;
  size_t total = (size_t)BB * 50 * NN;
  if (i >= total) return;
  int n = (int)(i % NN);
  size_t bc = i / NN;
  int c = (int)(bc % 50);
  int b = (int)(bc / 50);
  out[i] = Y[((size_t)b * NN + n) * 64 + c] + bias[c];
}

// ---------------- host orchestration ----------------
struct LayerW { int wi; int cout; int cin; };
static const LayerW LWS[23] = {
  {1, 32, 6},   {4, 32, 32},  {7, 64, 32},        // conv1
  {10, 64, 67}, {13, 64, 64}, {16, 128, 64},      // conv2
  {19, 128, 131}, {22, 128, 128}, {25, 256, 128}, // conv3
  {28, 256, 259}, {31, 256, 256}, {34, 512, 256}, // conv4
  {37, 256, 768}, {40, 256, 256},                 // dconv1
  {43, 256, 384}, {46, 256, 256},                 // dconv2
  {49, 256, 320}, {52, 128, 256},                 // dconv3
  {55, 128, 128}, {58, 128, 128},                 // dconv4
  {61, 128, 128}, {64, 128, 128},                 // mlp
  {67, 50, 128},                                  // fc (coutPad 64)
};

extern "C" void kernel_launch(void* const* d_in, const int* in_sizes, int n_in,
                              void* d_out, int out_size, void* d_ws, size_t ws_size,
                              hipStream_t stream) {
  (void)in_sizes; (void)n_in; (void)out_size; (void)ws_size;
  const float* x = (const float*)d_in[0];
  float* ws = (float*)d_ws;

  float* P  = ws;                 // B*N*3     =   98304
  float* F0 = P  + 98304;         // B*N*3     =   98304
  float* F1 = F0 + 98304;         // 4*2048*64 =  524288
  float* F2 = F1 + 524288;        // 4*512*128 =  262144
  float* F3 = F2 + 262144;        // 4*128*256 =  131072
  float* F4 = F3 + 131072;        // 4*32*512  =   65536
  float* U3 = F4 + 65536;         // 4*128*256 =  131072
  float* U2 = U3 + 131072;        // 4*512*256 =  524288
  float* U1 = U2 + 524288;        // 4*2048*128= 1048576
  float* U0 = U1 + 1048576;       // 4*8192*128= 4194304
  float* HA = U0 + 4194304;       // 16777216 (262144x64 max)
  float* HB = HA + 16777216;      // 16777216
  float* STATS = HB + 16777216;   // 1024
  float* D2 = STATS + 1024;       // 524288
  int*   IDX = (int*)(D2 + 524288);          // 524288 ints
  float* WARENA = D2 + 524288 + 524288;      // ~1M floats of packed weights

  // pack all weights (padded, zero-filled)
  float* wp[23];
  size_t woff = 0;
  for (int li = 0; li < 23; li++) {
    int coutPad = (LWS[li].cout + 15) & ~15;
    int cinPad  = (LWS[li].cin + 3) & ~3;
    wp[li] = WARENA + woff;
    woff += (size_t)coutPad * cinPad;
    size_t tot = (size_t)coutPad * cinPad;
    k_pack_w<<<(unsigned)((tot + 255) / 256), 256, 0, stream>>>(
        (const float*)d_in[LWS[li].wi], wp[li], LWS[li].cout, LWS[li].cin, coutPad, cinPad);
  }

  k_split<<<(BB * NN + 255) / 256, 256, 0, stream>>>(x, P, F0);

  auto mlp_layer = [&](const float* H, int li, float* Y, int rows) {
    int cout = LWS[li].cout;
    int cinPad = (LWS[li].cin + 3) & ~3;
    int tiles = (rows / 16) * (cout / 32);   // 16x32 strips; all couts are multiples of 32
    dim3 blk(32, 4);
    k_gemm_wmma<<<(tiles + 3) / 4, blk, 0, stream>>>(H, wp[li], Y, rows, cinPad, cout);
    k_zero<<<(2 * cout + 255) / 256, 256, 0, stream>>>(STATS, 2 * cout);
    size_t total = (size_t)rows * cout;
    int rb = (int)((total + 255) / 256); if (rb > 512) rb = 512;
    k_reduce_stats<<<rb, 256, 0, stream>>>(Y, rows, cout, STATS);
    k_bn_relu<<<(unsigned)((total + 255) / 256), 256, 0, stream>>>(
        Y, STATS, (const float*)d_in[LWS[li].wi + 1], (const float*)d_in[LWS[li].wi + 2], rows, cout);
  };

  auto conv_stage = [&](int qAbs, int Mq, int pAbs, int Np, const float* feat, int featC,
                        float radius, int li0, float* Fout) {
    int nq = BB * Mq;
    k_knn<32><<<(nq + 63) / 64, 64, 0, stream>>>(P, qAbs, Mq, pAbs, Np, radius * radius, IDX, (float*)nullptr);
    int rows = nq * 32;
    int cinPad0 = (LWS[li0].cin + 3) & ~3;
    size_t tot = (size_t)rows * cinPad0;
    k_build_conv_h<<<(unsigned)((tot + 255) / 256), 256, 0, stream>>>(
        P, feat, featC, Np, qAbs, Mq, pAbs, IDX, 32, HA, cinPad0);
    mlp_layer(HA, li0, HB, rows);
    mlp_layer(HB, li0 + 1, HA, rows);
    mlp_layer(HA, li0 + 2, HB, rows);
    int cout = LWS[li0 + 2].cout;
    size_t pt = (size_t)nq * cout;
    k_maxpool<<<(unsigned)((pt + 255) / 256), 256, 0, stream>>>(HB, Fout, nq, 32, cout);
  };

  auto deconv_stage = [&](int dAbs, int Md, int cAbs, int Mc, const float* fc, int Cc,
                          const float* fd, int Cd, int li0, float* Uout) {
    int nq = BB * Md;
    k_knn<16><<<(nq + 63) / 64, 64, 0, stream>>>(P, dAbs, Md, cAbs, Mc, -1.f, IDX, D2);
    int rows = nq;
    int cinPad0 = (LWS[li0].cin + 3) & ~3;
    size_t tot = (size_t)rows * cinPad0;
    k_build_deconv_h<<<(unsigned)((tot + 255) / 256), 256, 0, stream>>>(
        fc, Cc, Mc, fd, Cd, Md, IDX, D2, HA, cinPad0);
    mlp_layer(HA, li0, HB, rows);
    mlp_layer(HB, li0 + 1, Uout, rows);
  };

  // encoder
  conv_stage(4,   2048, 1,  8192, F0, 3,   0.1f, 0,  F1);
  conv_stage(16,  512,  4,  2048, F1, 64,  0.2f, 3,  F2);
  conv_stage(64,  128,  16, 512,  F2, 128, 0.4f, 6,  F3);
  conv_stage(256, 32,   64, 128,  F3, 256, 0.8f, 9,  F4);
  // decoder
  deconv_stage(64, 128,  256, 32,   F4, 512, F3, 256, 12, U3);
  deconv_stage(16, 512,  64,  128,  U3, 256, F2, 128, 14, U2);
  deconv_stage(4,  2048, 16,  512,  U2, 256, F1, 64,  16, U1);
  deconv_stage(1,  8192, 4,   2048, U1, 128, (const float*)nullptr, 0, 18, U0);
  // head
  mlp_layer(U0, 20, HA, BB * NN);
  mlp_layer(HA, 21, HB, BB * NN);
  {
    int rows = BB * NN, cinPad = 128, coutPad = 64;
    int tiles = (rows / 16) * (coutPad / 32);
    dim3 blk(32, 4);
    k_gemm_wmma<<<(tiles + 3) / 4, blk, 0, stream>>>(HB, wp[22], HA, rows, cinPad, coutPad);
    size_t tot = (size_t)BB * 50 * NN;
    k_fc_out<<<(unsigned)((tot + 255) / 256), 256, 0, stream>>>(HA, (const float*)d_in[68], (float*)d_out);
  }
}